// FeatureClusteringLoss_53137335386660
// MI455X (gfx1250) — compile-verified
//
#include <hip/hip_runtime.h>
#include <hip/hip_bf16.h>

// FeatureClusteringLoss on MI455X (gfx1250).
// Memory-bound: 256MB feature stream -> ~11us floor @ 23.3 TB/s.
// sq(n) = ||f_n||^2 - 2*f_n.p_{l_n} + ||p_{l_n}||^2 ; the dot-product term is
// computed as a 16x32 GEMM tile per wave with V_WMMA_F32_16X16X4_F32.

typedef __attribute__((ext_vector_type(2))) float v2f;
typedef __attribute__((ext_vector_type(8))) float v8f;

#define NUM_CLASSES 21
#define C_DIM       256
#define HW          16384            // 128*128
#define BATCH       16
#define PIX_TOTAL   (BATCH * HW)     // 262144
#define TILE        16
#define WAVES_PB    8
#define THREADS     (WAVES_PB * 32)
#define PROTO_PITCH 257              // 256 + 1 pad -> conflict-free LDS
#define G_PITCH     33               // 32 + 1 pad

__global__ __launch_bounds__(THREADS)
void fcl_main(const float* __restrict__ features,
              const int*   __restrict__ labels,
              const float* __restrict__ prototypes,
              float* __restrict__ gSse,
              float* __restrict__ gCnt)
{
    __shared__ float protoLds[32 * PROTO_PITCH];            // 32.9 KB
    __shared__ float gLds[WAVES_PB * 16 * G_PITCH];         // 16.9 KB
    __shared__ float psq[32];
    __shared__ float sseLds[NUM_CLASSES];
    __shared__ float cntLds[NUM_CLASSES];

    const int tid = threadIdx.x;

    // Stage prototypes into LDS (padded pitch), zero rows 21..31.
    for (int idx = tid; idx < NUM_CLASSES * C_DIM; idx += THREADS) {
        int k = idx >> 8, c = idx & 255;
        protoLds[k * PROTO_PITCH + c] = prototypes[idx];
    }
    for (int idx = tid; idx < (32 - NUM_CLASSES) * C_DIM; idx += THREADS) {
        int k = NUM_CLASSES + (idx >> 8), c = idx & 255;
        protoLds[k * PROTO_PITCH + c] = 0.0f;
    }
    if (tid < NUM_CLASSES) { sseLds[tid] = 0.0f; cntLds[tid] = 0.0f; }
    __syncthreads();

    // Per-class ||p||^2 (tiny: 21*256 mults per block).
    if (tid < 32) {
        float s = 0.0f;
        if (tid < NUM_CLASSES) {
            for (int c = 0; c < C_DIM; ++c) {
                float p = protoLds[tid * PROTO_PITCH + c];
                s = fmaf(p, p, s);
            }
        }
        psq[tid] = s;
    }
    __syncthreads();

    const int lane = tid & 31;
    const int wave = tid >> 5;
    const int half = lane >> 4;      // 0: K={0,1}, 1: K={2,3}
    const int m    = lane & 15;      // pixel row (A) / class col (B)

    const int tileId = blockIdx.x * WAVES_PB + wave;   // < 16384
    const int bIdx   = tileId >> 10;                   // tiles per image = 1024
    const int hw0    = (tileId & 1023) << 4;

    const float* fBase = features + (size_t)bIdx * C_DIM * HW + hw0 + m;

    float sqv    = 0.0f;
    int   lbl    = 0;
    bool  active = false;

#if __has_builtin(__builtin_amdgcn_wmma_f32_16x16x4_f32)
    v8f acc0 = {};   // classes 0..15
    v8f acc1 = {};   // classes 16..31 (21..31 are zero-padded)
    float fsq = 0.0f;

    for (int c0 = 0; c0 < C_DIM; c0 += 4) {
        const int ch = c0 + 2 * half;
        v2f a;
        a.x = fBase[(size_t)ch * HW];            // coalesced 64B per 16 lanes
        a.y = fBase[(size_t)(ch + 1) * HW];
        fsq = fmaf(a.x, a.x, fsq);
        fsq = fmaf(a.y, a.y, fsq);

        v2f b0, b1;
        b0.x = protoLds[m * PROTO_PITCH + ch];
        b0.y = protoLds[m * PROTO_PITCH + ch + 1];
        b1.x = protoLds[(16 + m) * PROTO_PITCH + ch];
        b1.y = protoLds[(16 + m) * PROTO_PITCH + ch + 1];

        acc0 = __builtin_amdgcn_wmma_f32_16x16x4_f32(false, a, false, b0,
                                                     (short)0, acc0, false, false);
        acc1 = __builtin_amdgcn_wmma_f32_16x16x4_f32(false, a, false, b1,
                                                     (short)0, acc1, false, false);
    }

    // Pixel m's channels are split between lane m (K%4 in {0,1}) and lane m+16.
    const float fsqT = fsq + __shfl_xor(fsq, 16, 32);

    // Spill D tile (16 pixels x 32 classes) to LDS for the label gather.
    float* gw = &gLds[wave * 16 * G_PITCH];
#pragma unroll
    for (int v = 0; v < 8; ++v) {
        const int row = v + 8 * half;            // D layout: VGPR v -> M=v / v+8
        gw[row * G_PITCH + m]      = acc0[v];
        gw[row * G_PITCH + 16 + m] = acc1[v];
    }
    __syncthreads();

    if (lane < 16) {
        const int n = bIdx * HW + hw0 + m;
        lbl = labels[n];
        if (lbl >= 0 && lbl < NUM_CLASSES) {     // 255 => excluded (ignore class)
            const float g = gw[m * G_PITCH + lbl];
            sqv = fsqT - 2.0f * g + psq[lbl];
            active = true;
        }
    }
#else
    // Scalar fallback (non-gfx1250 toolchains only).
    if (lane < 16) {
        const int n = bIdx * HW + hw0 + m;
        lbl = labels[n];
        if (lbl >= 0 && lbl < NUM_CLASSES) {
            float s = 0.0f;
            for (int c = 0; c < C_DIM; ++c) {
                float d = fBase[(size_t)c * HW] - protoLds[lbl * PROTO_PITCH + c];
                s = fmaf(d, d, s);
            }
            sqv = s;
            active = true;
        }
    }
    __syncthreads();
#endif

    // Per-class accumulation: LDS float atomics (ds_add_f32), then one global
    // atomic per class per block.
    if (active) {
        atomicAdd(&sseLds[lbl], sqv);
        atomicAdd(&cntLds[lbl], 1.0f);
    }
    __syncthreads();
    if (tid < NUM_CLASSES) {
        atomicAdd(&gSse[tid], sseLds[tid]);
        atomicAdd(&gCnt[tid], cntLds[tid]);
    }
}

__global__ void fcl_finalize(const float* __restrict__ gSse,
                             const float* __restrict__ gCnt,
                             float* __restrict__ out)
{
    if (threadIdx.x == 0 && blockIdx.x == 0) {
        float sum = 0.0f, np = 0.0f;
        for (int k = 0; k < NUM_CLASSES; ++k) {
            const float c = gCnt[k];
            if (c > 0.0f) {
                sum += gSse[k] / (c * (float)C_DIM);
                np  += 1.0f;
            }
        }
        out[0] = sum / fmaxf(np, 1.0f);   // FACTOR == 1.0
    }
}

extern "C" void kernel_launch(void* const* d_in, const int* in_sizes, int n_in,
                              void* d_out, int out_size, void* d_ws, size_t ws_size,
                              hipStream_t stream) {
    const float* features   = (const float*)d_in[0];
    const int*   labels     = (const int*)d_in[1];
    const float* prototypes = (const float*)d_in[2];
    float*       out        = (float*)d_out;

    float* gSse = (float*)d_ws;
    float* gCnt = gSse + NUM_CLASSES;

    // Zero the global per-class accumulators every launch (graph-capturable).
    hipMemsetAsync(d_ws, 0, 2 * NUM_CLASSES * sizeof(float), stream);

    const int blocks = PIX_TOTAL / (TILE * WAVES_PB);   // 2048
    fcl_main<<<blocks, THREADS, 0, stream>>>(features, labels, prototypes, gSse, gCnt);
    fcl_finalize<<<1, 32, 0, stream>>>(gSse, gCnt, out);
}